// EmaSquad_37203006717996
// MI455X (gfx1250) — compile-verified
//
#include <hip/hip_runtime.h>
#include <math.h>

typedef __attribute__((ext_vector_type(2))) float v2f;
typedef __attribute__((ext_vector_type(8))) float v8f;

#define BATCH 512
#define LSEQ  512
#define FIN   64
#define MCH   256
#define NEXP  4
#define KW    5
#define HID   512
#define CAP   384
#define NP0   29
#define NP1   13
#define NP2   5

__device__ __forceinline__ float gelu_exact(float x) {
    return 0.5f * x * (1.0f + erff(x * 0.7071067811865475f));
}

__device__ __forceinline__ v8f wmma_f32(v2f a, v2f b, v8f c) {
    // D = A(16x4,f32) * B(4x16,f32) + C(16x16,f32)
    return __builtin_amdgcn_wmma_f32_16x16x4_f32(false, a, false, b, (short)0, c, false, false);
}

// ---------------- conv input layer: x(512,512,64) -> h0(512,29,256) ----------------
// position p (0..28) corresponds to time L-29+p; window = x[b, L-33+p .. L-29+p, :]
// (320 contiguous floats). GEMM (512 x 320) @ (320 x 256) per position.
__global__ void conv0_kernel(const float* __restrict__ x,
                             const float* __restrict__ w,    // (256,64,5)
                             const float* __restrict__ bias, // (256)
                             float* __restrict__ h0) {
    int wid  = (blockIdx.x * blockDim.x + threadIdx.x) >> 5;
    int lane = threadIdx.x & 31;
    int p    = wid / 512;
    int rem  = wid % 512;
    int b0   = (rem >> 4) * 16;
    int n0   = (rem & 15) * 16;
    int half = lane >> 4;
    int mr   = lane & 15;

    const float* arow = x + (size_t)(b0 + mr) * (LSEQ * FIN) + (size_t)(LSEQ - 33 + p) * FIN;
    const float* bcol = w + (size_t)(n0 + mr) * (FIN * KW);

    v8f acc = {};
    for (int k0 = 0; k0 < FIN * KW; k0 += 4) {
        int kk = k0 + 2 * half;
        v2f a = *(const v2f*)(arow + kk);
        int j0 = kk >> 6, c0 = kk & 63;
        int j1 = (kk + 1) >> 6, c1 = (kk + 1) & 63;
        v2f bb;
        bb.x = bcol[c0 * KW + j0];
        bb.y = bcol[c1 * KW + j1];
        acc = wmma_f32(a, bb, acc);
    }
    float bv = bias[n0 + mr];
    for (int i = 0; i < 8; ++i) {
        int b = b0 + i + 8 * half;
        h0[((size_t)b * NP0 + p) * MCH + n0 + mr] = acc[i] + bv;
    }
}

// ---------------- residual dilated block (generic): hout[r] = hin[2r+4] + gelu(conv) ----
// window rows = hin rows [2r .. 2r+4] -> 1280 contiguous floats per batch.
__global__ void convblk_kernel(const float* __restrict__ hin,
                               const float* __restrict__ w,    // (256,256,5)
                               const float* __restrict__ bias, // (256)
                               float* __restrict__ hout,
                               int nrin, int nrout) {
    int wid  = (blockIdx.x * blockDim.x + threadIdx.x) >> 5;
    int lane = threadIdx.x & 31;
    int r    = wid / 512;
    int rem  = wid % 512;
    int b0   = (rem >> 4) * 16;
    int n0   = (rem & 15) * 16;
    int half = lane >> 4;
    int mr   = lane & 15;

    const float* arow = hin + ((size_t)(b0 + mr) * nrin + 2 * r) * MCH;
    const float* bcol = w + (size_t)(n0 + mr) * (MCH * KW);

    v8f acc = {};
    for (int k0 = 0; k0 < MCH * KW; k0 += 4) {
        int kk = k0 + 2 * half;
        v2f a = *(const v2f*)(arow + kk);
        int j0 = kk >> 8, c0 = kk & 255;
        int j1 = (kk + 1) >> 8, c1 = (kk + 1) & 255;
        v2f bb;
        bb.x = bcol[c0 * KW + j0];
        bb.y = bcol[c1 * KW + j1];
        acc = wmma_f32(a, bb, acc);
    }
    float bv = bias[n0 + mr];
    for (int i = 0; i < 8; ++i) {
        int b = b0 + i + 8 * half;
        float res = hin[((size_t)b * nrin + 2 * r + 4) * MCH + n0 + mr];
        hout[((size_t)b * nrout + r) * MCH + n0 + mr] = res + gelu_exact(acc[i] + bv);
    }
}

// ---------------- layernorm over M=256 per token ----------------
__global__ void ln_kernel(const float* __restrict__ hraw, const float* __restrict__ g,
                          const float* __restrict__ bln, float* __restrict__ tok) {
    __shared__ float red[256];
    int t = blockIdx.x;
    int m = threadIdx.x;
    float v = hraw[(size_t)t * MCH + m];
    red[m] = v; __syncthreads();
    for (int s = 128; s > 0; s >>= 1) { if (m < s) red[m] += red[m + s]; __syncthreads(); }
    float mu = red[0] * (1.0f / MCH); __syncthreads();
    float d = v - mu;
    red[m] = d * d; __syncthreads();
    for (int s = 128; s > 0; s >>= 1) { if (m < s) red[m] += red[m + s]; __syncthreads(); }
    float var = red[0] * (1.0f / MCH);
    tok[(size_t)t * MCH + m] = d * rsqrtf(var + 1e-5f) * g[m] + bln[m];
}

// ---------------- top-2 routing with capacity (ordered scan, 1 block) ----------------
__global__ void route_kernel(const float* __restrict__ tok, const float* __restrict__ wg,
                             int* __restrict__ me1, int* __restrict__ ml1, int* __restrict__ mk1,
                             int* __restrict__ me2, int* __restrict__ ml2, int* __restrict__ mk2,
                             float* __restrict__ mg1, float* __restrict__ mg2) {
    __shared__ int se1[BATCH];
    __shared__ int se2[BATCH];
    __shared__ int cnt1[NEXP];
    int t = threadIdx.x;
    float sc[4] = {0.f, 0.f, 0.f, 0.f};
    const float* tr = tok + (size_t)t * MCH;
    for (int m = 0; m < MCH; ++m) {
        float tv = tr[m];
        const float* wr = wg + m * NEXP;
        sc[0] += tv * wr[0]; sc[1] += tv * wr[1];
        sc[2] += tv * wr[2]; sc[3] += tv * wr[3];
    }
    float mx = fmaxf(fmaxf(sc[0], sc[1]), fmaxf(sc[2], sc[3]));
    float p[4]; float denom = 0.f;
    for (int e = 0; e < 4; ++e) { p[e] = expf(sc[e] - mx); denom += p[e]; }
    for (int e = 0; e < 4; ++e) p[e] /= denom;
    int e1 = 0; float g1 = p[0];
    for (int e = 1; e < 4; ++e) if (p[e] > g1) { g1 = p[e]; e1 = e; }
    int e2 = -1; float g2 = -1.0f;
    for (int e = 0; e < 4; ++e) if (e != e1 && p[e] > g2) { g2 = p[e]; e2 = e; }
    se1[t] = e1; se2[t] = e2;
    __syncthreads();
    int loc1 = 0;
    for (int tp = 0; tp < t; ++tp) loc1 += (se1[tp] == e1);
    if (t < NEXP) {
        int c = 0;
        for (int tp = 0; tp < BATCH; ++tp) c += (se1[tp] == t);
        cnt1[t] = c;
    }
    __syncthreads();
    int loc2 = cnt1[e2];
    for (int tp = 0; tp < t; ++tp) loc2 += (se2[tp] == e2);
    int k1 = loc1 < CAP, k2 = loc2 < CAP;
    me1[t] = e1; ml1[t] = k1 ? loc1 : (CAP - 1); mk1[t] = k1; mg1[t] = k1 ? g1 : 0.0f;
    me2[t] = e2; ml2[t] = k2 ? loc2 : (CAP - 1); mk2[t] = k2; mg2[t] = k2 ? g2 : 0.0f;
}

__global__ void zero_kernel(float* __restrict__ p, int n) {
    int i = blockIdx.x * 256 + threadIdx.x;
    if (i < n) p[i] = 0.0f;
}

__global__ void dispatch_kernel(const float* __restrict__ tok,
                                const int* __restrict__ me1, const int* __restrict__ ml1,
                                const int* __restrict__ mk1,
                                const int* __restrict__ me2, const int* __restrict__ ml2,
                                const int* __restrict__ mk2, float* __restrict__ disp) {
    int t = blockIdx.x, m = threadIdx.x;
    float v = tok[(size_t)t * MCH + m];
    if (mk1[t]) disp[((size_t)me1[t] * CAP + ml1[t]) * MCH + m] = v;
    if (mk2[t]) disp[((size_t)me2[t] * CAP + ml2[t]) * MCH + m] = v;
}

// ---------------- per-expert FFN GEMM: (CAP x Kt) @ (Kt x N) + bias [gelu] ----------------
__global__ void ffn_kernel(const float* __restrict__ A,    // (E, CAP, Kt)
                           const float* __restrict__ W,    // (E, Kt, N)
                           const float* __restrict__ bias, // (E, N)
                           float* __restrict__ Out,        // (E, CAP, N)
                           int N, int Kt, int doGelu) {
    int wid  = (blockIdx.x * blockDim.x + threadIdx.x) >> 5;
    int lane = threadIdx.x & 31;
    int ntiles = N >> 4;
    int tilesPerE = (CAP >> 4) * ntiles;
    int e   = wid / tilesPerE;
    int rem = wid % tilesPerE;
    int r0  = (rem / ntiles) * 16;
    int n0  = (rem % ntiles) * 16;
    int half = lane >> 4;
    int mr   = lane & 15;

    const float* arow  = A + ((size_t)e * CAP + r0 + mr) * Kt;
    const float* bbase = W + (size_t)e * Kt * N + n0 + mr;
    v8f acc = {};
    for (int k0 = 0; k0 < Kt; k0 += 4) {
        int kk = k0 + 2 * half;
        v2f a = *(const v2f*)(arow + kk);
        v2f bb;
        bb.x = bbase[(size_t)kk * N];
        bb.y = bbase[(size_t)(kk + 1) * N];
        acc = wmma_f32(a, bb, acc);
    }
    float bv = bias[e * N + n0 + mr];
    for (int i = 0; i < 8; ++i) {
        int row = r0 + i + 8 * half;
        float v = acc[i] + bv;
        if (doGelu) v = gelu_exact(v);
        Out[((size_t)e * CAP + row) * N + n0 + mr] = v;
    }
}

__global__ void combine_kernel(const float* __restrict__ y,
                               const int* __restrict__ me1, const int* __restrict__ ml1,
                               const int* __restrict__ me2, const int* __restrict__ ml2,
                               const float* __restrict__ mg1, const float* __restrict__ mg2,
                               float* __restrict__ out) {
    int t = blockIdx.x, m = threadIdx.x;
    float v1 = y[((size_t)me1[t] * CAP + ml1[t]) * MCH + m] * mg1[t];
    float v2 = y[((size_t)me2[t] * CAP + ml2[t]) * MCH + m] * mg2[t];
    out[(size_t)t * MCH + m] = v1 + v2;
}

extern "C" void kernel_launch(void* const* d_in, const int* in_sizes, int n_in,
                              void* d_out, int out_size, void* d_ws, size_t ws_size,
                              hipStream_t stream) {
    const float* x   = (const float*)d_in[0];
    const float* wi  = (const float*)d_in[1];
    const float* bi  = (const float*)d_in[2];
    const float* w0  = (const float*)d_in[3];
    const float* b0c = (const float*)d_in[4];
    const float* w1c = (const float*)d_in[5];
    const float* b1c = (const float*)d_in[6];
    const float* w2c = (const float*)d_in[7];
    const float* b2c = (const float*)d_in[8];
    const float* lng = (const float*)d_in[9];
    const float* lnb = (const float*)d_in[10];
    const float* wg  = (const float*)d_in[11];
    const float* w1  = (const float*)d_in[12];
    const float* bb1 = (const float*)d_in[13];
    const float* w2  = (const float*)d_in[14];
    const float* bb2 = (const float*)d_in[15];
    float* out = (float*)d_out;

    float* ws   = (float*)d_ws;
    float* h0   = ws;                              // 512*29*256
    float* h1   = h0   + (size_t)512 * 29 * 256;   // 512*13*256
    float* h2   = h1   + (size_t)512 * 13 * 256;   // 512*5*256
    float* traw = h2   + (size_t)512 * 5 * 256;    // 512*256
    float* tok  = traw + (size_t)512 * 256;        // 512*256
    float* disp = tok  + (size_t)512 * 256;        // 4*384*256
    float* hmid = disp + (size_t)4 * 384 * 256;    // 4*384*512
    float* yws  = hmid + (size_t)4 * 384 * 512;    // 4*384*256
    int*   meta = (int*)(yws + (size_t)4 * 384 * 256);
    int* me1 = meta;        int* ml1 = me1 + 512;  int* mk1 = ml1 + 512;
    int* me2 = mk1 + 512;   int* ml2 = me2 + 512;  int* mk2 = ml2 + 512;
    float* mg1 = (float*)(mk2 + 512);
    float* mg2 = mg1 + 512;

    // conv stack on dependency cone of the terminal timestep
    conv0_kernel<<<(NP0 * 512) / 8, 256, 0, stream>>>(x, wi, bi, h0);
    convblk_kernel<<<(NP1 * 512) / 8, 256, 0, stream>>>(h0, w0, b0c, h1, NP0, NP1);
    convblk_kernel<<<(NP2 * 512) / 8, 256, 0, stream>>>(h1, w1c, b1c, h2, NP1, NP2);
    convblk_kernel<<<(1 * 512) / 8, 256, 0, stream>>>(h2, w2c, b2c, traw, NP2, 1);
    // layernorm -> routing -> dispatch
    ln_kernel<<<512, 256, 0, stream>>>(traw, lng, lnb, tok);
    route_kernel<<<1, 512, 0, stream>>>(tok, wg, me1, ml1, mk1, me2, ml2, mk2, mg1, mg2);
    zero_kernel<<<(4 * 384 * 256 + 255) / 256, 256, 0, stream>>>(disp, 4 * 384 * 256);
    dispatch_kernel<<<512, 256, 0, stream>>>(tok, me1, ml1, mk1, me2, ml2, mk2, disp);
    // expert FFN (WMMA) and combine
    ffn_kernel<<<(4 * 24 * 32) / 8, 256, 0, stream>>>(disp, w1, bb1, hmid, 512, 256, 1);
    ffn_kernel<<<(4 * 24 * 16) / 8, 256, 0, stream>>>(hmid, w2, bb2, yws, 256, 512, 0);
    combine_kernel<<<512, 256, 0, stream>>>(yws, me1, ml1, me2, ml2, mg1, mg2, out);
}